// GraphConvolution_24953759990541
// MI455X (gfx1250) — compile-verified
//
#include <hip/hip_runtime.h>
#include <hip/hip_bf16.h>

typedef __attribute__((ext_vector_type(2))) float v2f;
typedef __attribute__((ext_vector_type(8))) float v8f;

#define NNODES 50000
#define NBATCH 4
#define DIN    128
#define DOUT   64

// ---------------------------------------------------------------------------
// pre[r, :] = x[r, :] @ W   for r in [0, nrows), using V_WMMA_F32_16X16X4_F32.
// Each wave computes a 16-row x 64-col stripe: 4 accumulators, K-loop of 32.
// A fragment (16x4 f32): lanes 0-15 hold rows M=0..15, lane group (lane>>4)
// selects K pair {0,1} or {2,3}; v0=K even, v1=K odd.
// B fragment (4x16 f32): lanes 0-15 hold cols N=0..15, same K-pair scheme.
// C/D (16x16 f32): VGPR v, lanes 0-15 -> M=v, lanes 16-31 -> M=v+8, N=lane&15.
// ---------------------------------------------------------------------------
__global__ void gcn_gemm_wmma(const float* __restrict__ x,
                              const float* __restrict__ w,
                              float* __restrict__ pre,
                              int nrows) {
  const int wave = (int)((blockIdx.x * blockDim.x + threadIdx.x) >> 5);
  const int lane = (int)(threadIdx.x & 31);
  const int ntiles = nrows >> 4;
  if (wave >= ntiles) return;

  const int row0 = wave << 4;
  const int m    = lane & 15;          // row within tile (A) / col within tile (B)
  const int kh   = (lane >> 4) << 1;   // K sub-pair: 0 or 2

  const float* __restrict__ xrow = x + (size_t)(row0 + m) * DIN;

  v8f acc0 = {}, acc1 = {}, acc2 = {}, acc3 = {};

  for (int k = 0; k < DIN; k += 4) {
    v2f a;
    a.x = xrow[k + kh];
    a.y = xrow[k + kh + 1];

    const float* __restrict__ wr0 = w + (size_t)(k + kh) * DOUT;
    const float* __restrict__ wr1 = wr0 + DOUT;

    v2f b0, b1, b2, b3;
    b0.x = wr0[m +  0]; b0.y = wr1[m +  0];
    b1.x = wr0[m + 16]; b1.y = wr1[m + 16];
    b2.x = wr0[m + 32]; b2.y = wr1[m + 32];
    b3.x = wr0[m + 48]; b3.y = wr1[m + 48];

    acc0 = __builtin_amdgcn_wmma_f32_16x16x4_f32(false, a, false, b0, (short)0, acc0, false, false);
    acc1 = __builtin_amdgcn_wmma_f32_16x16x4_f32(false, a, false, b1, (short)0, acc1, false, false);
    acc2 = __builtin_amdgcn_wmma_f32_16x16x4_f32(false, a, false, b2, (short)0, acc2, false, false);
    acc3 = __builtin_amdgcn_wmma_f32_16x16x4_f32(false, a, false, b3, (short)0, acc3, false, false);
  }

  // Store 16x64 stripe.
  const int nc    = lane & 15;
  const int mbase = (lane >> 4) << 3;  // 0 or 8
  float* __restrict__ outp = pre + (size_t)row0 * DOUT;
#pragma unroll
  for (int v = 0; v < 8; ++v) {
    const size_t mm = (size_t)(mbase + v) * DOUT;
    outp[mm + nc +  0] = acc0[v];
    outp[mm + nc + 16] = acc1[v];
    outp[mm + nc + 32] = acc2[v];
    outp[mm + nc + 48] = acc3[v];
  }
}

// ---------------------------------------------------------------------------
// For each (batch b = blockIdx.y, edge e = wave): out[b, rows[e], :] +=
// vals[e] * pre[b, cols[e], :].  Lane handles 2 features (float2 gather,
// two global_atomic_add_f32).  pre + out both fit in the 192 MB L2, so the
// random traffic stays on-chip.
// ---------------------------------------------------------------------------
__global__ void gcn_scatter(const float* __restrict__ pre,
                            const float* __restrict__ evals,
                            const int* __restrict__ erows,
                            const int* __restrict__ ecols,
                            float* __restrict__ out,
                            int nedges) {
  const int e    = (int)((blockIdx.x * blockDim.x + threadIdx.x) >> 5);
  const int lane = (int)(threadIdx.x & 31);
  if (e >= nedges) return;
  const int b = (int)blockIdx.y;

  const int   c   = ecols[e];
  const int   r   = erows[e];
  const float val = evals[e];

  const float* __restrict__ src = pre + ((size_t)b * NNODES + c) * DOUT + lane * 2;
  float*       __restrict__ dst = out + ((size_t)b * NNODES + r) * DOUT + lane * 2;

  const float2 g = *(const float2*)src;
  unsafeAtomicAdd(dst + 0, g.x * val);
  unsafeAtomicAdd(dst + 1, g.y * val);
}

__global__ void gcn_zero(float4* __restrict__ p, int n4) {
  const int i = (int)(blockIdx.x * blockDim.x + threadIdx.x);
  if (i < n4) p[i] = make_float4(0.f, 0.f, 0.f, 0.f);
}

__global__ void gcn_relu(float4* __restrict__ p, int n4) {
  const int i = (int)(blockIdx.x * blockDim.x + threadIdx.x);
  if (i < n4) {
    float4 v = p[i];
    v.x = v.x > 0.f ? v.x : 0.f;
    v.y = v.y > 0.f ? v.y : 0.f;
    v.z = v.z > 0.f ? v.z : 0.f;
    v.w = v.w > 0.f ? v.w : 0.f;
    p[i] = v;
  }
}

extern "C" void kernel_launch(void* const* d_in, const int* in_sizes, int n_in,
                              void* d_out, int out_size, void* d_ws, size_t ws_size,
                              hipStream_t stream) {
  const float* x     = (const float*)d_in[0];   // [4, 50000, 128]
  const float* w     = (const float*)d_in[1];   // [128, 64]
  const float* evals = (const float*)d_in[2];   // [800000]
  const int*   erows = (const int*)  d_in[3];   // [800000]
  const int*   ecols = (const int*)  d_in[4];   // [800000]
  float*       out   = (float*)d_out;           // [4, 50000, 64]
  float*       pre   = (float*)d_ws;            // [4*50000, 64] scratch

  const int nedges = in_sizes[2];
  const int nrows  = in_sizes[0] / DIN;         // 4*50000 = 200000
  const int n4     = out_size / 4;              // float4 count

  // 1) zero accumulation target
  {
    const int threads = 256;
    const int blocks  = (n4 + threads - 1) / threads;
    gcn_zero<<<blocks, threads, 0, stream>>>((float4*)out, n4);
  }

  // 2) dense GEMM via fp32 WMMA: 8 waves/block, one 16x64 stripe per wave
  {
    const int threads  = 256;                   // 8 waves
    const int ntiles   = nrows / 16;            // 12500
    const int blocks   = (ntiles + 7) / 8;
    gcn_gemm_wmma<<<blocks, threads, 0, stream>>>(x, w, pre, nrows);
  }

  // 3) gather + scale + scatter-add (atomics), one wave per (edge, batch)
  {
    const int threads = 256;                    // 8 edges per block
    dim3 grid((nedges + 7) / 8, NBATCH, 1);
    gcn_scatter<<<grid, threads, 0, stream>>>(pre, evals, erows, ecols, out, nedges);
  }

  // 4) ReLU in place
  {
    const int threads = 256;
    const int blocks  = (n4 + threads - 1) / threads;
    gcn_relu<<<blocks, threads, 0, stream>>>((float4*)out, n4);
  }
}